// GLayer_57458072486300
// MI455X (gfx1250) — compile-verified
//
#include <hip/hip_runtime.h>
#include <math.h>

// ---------------------------------------------------------------------------
// GLayer forward for MI455X (gfx1250, wave32).
//   * One workgroup (256 threads = 8 waves) per batch element.
//   * G_herm built directly into LDS, padded 129 -> 144 with zeros.
//   * Cyclic parallel Jacobi eigensolver fully in LDS (A + V ~ 166 KB,
//     fits CDNA5's 320 KB/WGP LDS).
//   * Eigenvalue MLP (1->16->1) evaluated per eigenvalue in registers.
//   * Reconstruction V diag(v) V^T via V_WMMA_F32_16X16X4_F32 tiles.
// ---------------------------------------------------------------------------

typedef __attribute__((ext_vector_type(2))) float v2f;
typedef __attribute__((ext_vector_type(8))) float v8f;

#define DIM      129
#define MN       128
#define NPAD     144           // next multiple of 16 (9 WMMA tiles)
#define STR      145           // odd LDS row stride -> bank-conflict friendly
#define NPAIR    72            // NPAD/2 disjoint Jacobi pairs per rotation set
#define NROUND   143           // NPAD-1 rotation sets per sweep
#define NSWEEP   8             // fixed sweep count (deterministic)
#define NTHREADS 256
#define EPS      1e-8f

__device__ __forceinline__ float softplusf(float x) {
    return fmaxf(x, 0.0f) + log1pf(expf(-fabsf(x)));
}
__device__ __forceinline__ float sigmoidf(float x) {
    return 1.0f / (1.0f + expf(-x));
}

// round-robin tournament pairing: round r, pair index k -> (p,q), p<q
__device__ __forceinline__ void jacobi_pair(int r, int k, int* pp, int* qq) {
    int p, q;
    if (k == 0) { p = NPAD - 1; q = r; }
    else {
        p = (r + k) % NROUND;
        q = (r - k + NROUND) % NROUND;
    }
    if (p > q) { int t = p; p = q; q = t; }
    *pp = p; *qq = q;
}

extern "C" __global__ __launch_bounds__(NTHREADS)
void glayer_fused(const float* __restrict__ phi, const float* __restrict__ H,
                  const float* __restrict__ Z,
                  const float* __restrict__ lambda_p, const float* __restrict__ rho_p,
                  const float* __restrict__ thr_p,
                  const float* __restrict__ W1, const float* __restrict__ b1,
                  const float* __restrict__ W2, const float* __restrict__ b2,
                  float* __restrict__ out)
{
    extern __shared__ float lds[];
    float* A     = lds;                       // NPAD*STR: matrix, later V*diag(v)
    float* V     = lds + NPAD * STR;          // NPAD*STR: eigenvectors
    float* cbuf  = lds + 2 * NPAD * STR;      // NPAIR rotation cosines
    float* sbuf  = cbuf + NPAIR;              // NPAIR rotation sines
    float* vproc = sbuf + NPAIR;              // NPAD processed eigenvalues

    const int b   = blockIdx.x;
    const int tid = threadIdx.x;

    const float* Hb   = H   + (size_t)b * MN * MN;
    const float* Zb   = Z   + (size_t)b * DIM * DIM;
    const float* phib = phi + (size_t)b * MN;

    __builtin_prefetch((const void*)Hb, 0, 1);   // global_prefetch path
    __builtin_prefetch((const void*)Zb, 0, 1);

    // scalar parameters (cheap, recomputed per thread)
    const float lam     = softplusf(lambda_p[0]);
    const float lam_inv = 1.0f / (lam * lam + EPS);
    const float rden    = softplusf(rho_p[0]) + EPS;
    const float th      = sigmoidf(thr_p[0]);

    // ---- build symmetric padded matrix in LDS, V = I ----------------------
    for (int idx = tid; idx < NPAD * NPAD; idx += NTHREADS) {
        const int i = idx / NPAD, j = idx % NPAD;
        float val = 0.0f;
        if (i < DIM && j < DIM) {
            float bij = (i < MN) ? ((j < MN) ? Hb[i * MN + j] : phib[i])
                                 : ((j < MN) ? phib[j] : lam_inv);
            float bji = (j < MN) ? ((i < MN) ? Hb[j * MN + i] : phib[j])
                                 : ((i < MN) ? phib[i] : lam_inv);
            float zij = Zb[i * DIM + j];
            float zji = Zb[j * DIM + i];
            val = 0.5f * (bij + bji) - 0.5f * (zij + zji) / rden;
        }
        A[i * STR + j] = val;
        V[i * STR + j] = (i == j) ? 1.0f : 0.0f;
    }
    __syncthreads();

    // ---- cyclic parallel Jacobi: A' = J^T A J, V' = V J -------------------
    for (int sw = 0; sw < NSWEEP; ++sw) {
        for (int r = 0; r < NROUND; ++r) {
            // 1) rotation angles for the 72 disjoint pairs
            if (tid < NPAIR) {
                int p, q; jacobi_pair(r, tid, &p, &q);
                const float app = A[p * STR + p];
                const float aqq = A[q * STR + q];
                const float apq = A[p * STR + q];
                float c = 1.0f, s = 0.0f;
                if (fabsf(apq) > 1e-20f) {
                    const float tau = (aqq - app) / (2.0f * apq);
                    const float t   = copysignf(1.0f, tau) /
                                      (fabsf(tau) + sqrtf(1.0f + tau * tau));
                    c = rsqrtf(1.0f + t * t);
                    s = t * c;
                }
                cbuf[tid] = c; sbuf[tid] = s;
            }
            __syncthreads();
            // 2) row phase: B = J^T A
            for (int idx = tid; idx < NPAIR * NPAD; idx += NTHREADS) {
                const int pr = idx / NPAD, j = idx % NPAD;
                int p, q; jacobi_pair(r, pr, &p, &q);
                const float c = cbuf[pr], s = sbuf[pr];
                const float apj = A[p * STR + j];
                const float aqj = A[q * STR + j];
                A[p * STR + j] = c * apj - s * aqj;
                A[q * STR + j] = s * apj + c * aqj;
            }
            __syncthreads();
            // 3) column phase: A = B J, and V = V J
            for (int idx = tid; idx < NPAIR * NPAD; idx += NTHREADS) {
                const int pr = idx / NPAD, i = idx % NPAD;
                int p, q; jacobi_pair(r, pr, &p, &q);
                const float c = cbuf[pr], s = sbuf[pr];
                const float aip = A[i * STR + p];
                const float aiq = A[i * STR + q];
                A[i * STR + p] = c * aip - s * aiq;
                A[i * STR + q] = s * aip + c * aiq;
                const float vip = V[i * STR + p];
                const float viq = V[i * STR + q];
                V[i * STR + p] = c * vip - s * viq;
                V[i * STR + q] = s * vip + c * viq;
            }
            __syncthreads();
        }
    }

    // ---- eigenvalue projection (softplus gate * tiny MLP) -----------------
    for (int i = tid; i < NPAD; i += NTHREADS) {
        float vp = 0.0f;
        if (i < DIM) {
            const float ev   = A[i * STR + i];
            const float base = softplusf(ev - th);
            const float habs = fabsf(ev);
            float acc = b2[0];
            #pragma unroll
            for (int j = 0; j < 16; ++j)
                acc += fmaxf(habs * W1[j] + b1[j], 0.0f) * W2[j];
            vp = base * sigmoidf(acc);
        }
        vproc[i] = vp;
    }
    __syncthreads();

    // ---- A <- V * diag(vproc)  (A no longer needed) -----------------------
    for (int idx = tid; idx < NPAD * NPAD; idx += NTHREADS) {
        const int i = idx / NPAD, k = idx % NPAD;
        A[i * STR + k] = V[i * STR + k] * vproc[k];
    }
    __syncthreads();

    // ---- G = A * V^T via V_WMMA_F32_16X16X4_F32 ---------------------------
    // fp32 WMMA fragment layout (wave32):
    //   A (16x4): lane L, vgpr r  -> A[M = L%16][K = r + 2*(L/16)]
    //   B (4x16): lane L, vgpr r  -> B[K = r + 2*(L/16)][N = L%16]
    //             B = V^T, so B[k][n] = V[tile_j*16 + n][k]
    //   D (16x16): vgpr v, lane L -> D[M = v + 8*(L/16)][N = L%16]
    const int wave = tid >> 5;      // wave32: 8 waves per block
    const int lane = tid & 31;
    const int m    = lane & 15;
    const int half = lane >> 4;
    float* outb = out + (size_t)b * DIM * DIM;

    for (int t = wave; t < 81; t += 8) {          // wave-uniform -> EXEC all-1s
        const int ti = t / 9, tj = t % 9;
        v8f acc = {0.f, 0.f, 0.f, 0.f, 0.f, 0.f, 0.f, 0.f};
        const float* Arow = A + (ti * 16 + m) * STR;
        const float* Vrow = V + (tj * 16 + m) * STR;
        #pragma unroll 4
        for (int k = 0; k < NPAD / 4; ++k) {
            const int kk = k * 4 + 2 * half;
            v2f af; af.x = Arow[kk]; af.y = Arow[kk + 1];
            v2f bf; bf.x = Vrow[kk]; bf.y = Vrow[kk + 1];
            acc = __builtin_amdgcn_wmma_f32_16x16x4_f32(
                      false, af, false, bf, (short)0, acc, false, false);
        }
        #pragma unroll
        for (int v = 0; v < 8; ++v) {
            const int row = ti * 16 + v + 8 * half;
            const int col = tj * 16 + m;
            if (row < DIM && col < DIM)
                outb[row * DIM + col] = acc[v];
        }
    }
}

extern "C" void kernel_launch(void* const* d_in, const int* in_sizes, int n_in,
                              void* d_out, int out_size, void* d_ws, size_t ws_size,
                              hipStream_t stream) {
    const float* phi = (const float*)d_in[0];
    const float* H   = (const float*)d_in[1];
    const float* Z   = (const float*)d_in[2];
    const float* lam = (const float*)d_in[3];
    const float* rho = (const float*)d_in[4];
    const float* thr = (const float*)d_in[5];
    const float* W1  = (const float*)d_in[6];
    const float* b1  = (const float*)d_in[7];
    const float* W2  = (const float*)d_in[8];
    const float* b2  = (const float*)d_in[9];
    float* out = (float*)d_out;

    const int B = in_sizes[0] / MN;   // 1024
    const size_t lds_bytes =
        (size_t)(2 * NPAD * STR + 2 * NPAIR + NPAD) * sizeof(float); // ~168 KB

    glayer_fused<<<B, NTHREADS, lds_bytes, stream>>>(
        phi, H, Z, lam, rho, thr, W1, b1, W2, b2, out);
}